// myConv_11038065950911
// MI455X (gfx1250) — compile-verified
//
#include <hip/hip_runtime.h>
#include <hip/hip_bf16.h>

// CDNA5 / gfx1250: 3x3 single-kernel conv broadcast over all (b,c) planes.
// Per-wave 16x16 output tile = 15x V_WMMA_F32_16X16X4_F32:
//   for each vertical tap p: D += A_p(16x20) x B_p(20x16)
// A_p[i][k] = x[row+i+p-1][col+k-1] (LDS patch, halo-padded, all finite)
// B_p[k][j] = w[p][k-j] for 0<=k-j<=2 (banded Toeplitz; zero for k>=18, so
//             the A tail over-read needs only finiteness, no masking).
// B table built by SCATTER: each (m,p,q) maps to exactly one cell via
// kk=m+q -> (t,hi,s), so 144 threads write 144 weights, no select trees.

typedef __attribute__((ext_vector_type(2))) float v2f;
typedef __attribute__((ext_vector_type(4))) float f4;
typedef __attribute__((ext_vector_type(8))) float v8f;

#define PLANE 256
#define PROWS 18          // 16 + 2 halo rows
#define PSTRIDE 132       // 4r mod 64 banks -> conflict-free A-row access
#define BSTRIDE 34        // dwords per lane in B table: even (8B aligned), bank-clean

__global__ __launch_bounds__(256) void conv3x3_wmma_f32(
    const float* __restrict__ X,   // (B*C, 256, 256) planes
    const float* __restrict__ Wt,  // 9 weights, row-major 3x3
    float* __restrict__ Out)
{
    __shared__ float patch[PROWS * PSTRIDE];   // 9504 B
    __shared__ float btab[32 * BSTRIDE];       // 4352 B

    const int tid   = threadIdx.x;
    const int blk   = blockIdx.x;
    // 32 blocks per plane: 16 tile-rows x 2 column halves (128 cols each)
    const int plane = blk >> 5;
    const int rem   = blk & 31;
    const int trow  = rem >> 1;
    const int half  = rem & 1;
    const int colb  = half << 7;      // 0 or 128
    const int rowb  = trow << 4;      // tile-row * 16

    const float* Xp = X + (size_t)plane * (PLANE * PLANE);

    // ---- phase 0: zero the B table (272 b128 stores block-wide) ----
    for (int idx = tid; idx < (32 * BSTRIDE) / 4; idx += 256) {
        f4 z = {0.0f, 0.0f, 0.0f, 0.0f};
        *(f4*)&btab[idx * 4] = z;
    }
    __syncthreads();   // only the zero-stores are outstanding here (cheap)

    // ---- phase 1a: interior patch, 18 rows x 32 float4 (cols 1..128) ----
    for (int idx = tid; idx < PROWS * 32; idx += 256) {
        const int r   = idx >> 5;
        const int c4  = idx & 31;
        const int gr  = rowb - 1 + r;
        const bool ok = (unsigned)gr < (unsigned)PLANE;
        const int grc = ok ? gr : 0;                    // clamped: load always valid
        f4 v = *(const f4*)&Xp[grc * PLANE + colb + 4 * c4];
        float* dst = &patch[r * PSTRIDE + 1 + 4 * c4];
        dst[0] = ok ? v.x : 0.0f;
        dst[1] = ok ? v.y : 0.0f;
        dst[2] = ok ? v.z : 0.0f;
        dst[3] = ok ? v.w : 0.0f;
    }

    // ---- phase 1b: halo cols {0,129,130,131} x 18 rows (72 scalars) ----
    if (tid < 72) {
        const int r   = tid >> 2;
        const int hc  = tid & 3;
        const int c   = hc ? (128 + hc) : 0;  // 0,129,130,131
        const int gr  = rowb - 1 + r;
        const int gc  = colb - 1 + c;
        const bool ok = ((unsigned)gr < (unsigned)PLANE) & ((unsigned)gc < (unsigned)PLANE);
        const int ofs = ok ? (gr * PLANE + gc) : 0;     // clamped address
        float v = Xp[ofs];
        patch[r * PSTRIDE + c] = ok ? v : 0.0f;
    }

    // ---- phase 1c: scatter the 144 nonzero B-table entries ----
    // tid = pq*16 + m ; pq = 3p+q ; kk = m+q = 4t + 2hi + s (unique decomp)
    if (tid < 144) {
        const int m  = tid & 15;
        const int pq = tid >> 4;          // 0..8
        const int p  = pq / 3;
        const int q  = pq - 3 * p;
        const int kk = m + q;             // 0..17
        const int t  = kk >> 2;
        const int h2 = (kk >> 1) & 1;
        const int s  = kk & 1;
        btab[((h2 << 4) | m) * BSTRIDE + (p * 5 + t) * 2 + s] = Wt[pq];
    }

    __syncthreads();

    // ---- per-lane indices ----
    const int lane = tid & 31;
    const int wid  = tid >> 5;        // wave 0..7 -> 16-col sub-tile
    const int m    = lane & 15;       // A row / D col index
    const int hi   = lane >> 4;
    const int w16  = wid << 4;

    // ---- fetch B fragments from the shared table (b64 loads, no VALU) ----
    v2f bf[3][5];
    const v2f* bl = (const v2f*)&btab[lane * BSTRIDE];
#pragma unroll
    for (int p = 0; p < 3; ++p)
#pragma unroll
        for (int t = 0; t < 5; ++t)
            bf[p][t] = bl[p * 5 + t];

    // ---- 15 WMMAs: pure ds_load -> v_wmma chain ----
    v8f acc = {};
#pragma unroll
    for (int p = 0; p < 3; ++p) {
        const float* rowp = &patch[(m + p) * PSTRIDE + w16];
#pragma unroll
        for (int t = 0; t < 5; ++t) {
            const int koff = 4 * t + 2 * hi;
            v2f a;
            a.x = rowp[koff];          // k=18,19 over-read: initialized, finite,
            a.y = rowp[koff + 1];      // multiplied by B == 0
            acc = __builtin_amdgcn_wmma_f32_16x16x4_f32(
                false, a, false, bf[p][t], (short)0, acc, false, false);
        }
    }

    // ---- store D: VGPR v -> row (rowb + v + 8*hi), col (colb + w16 + m) ----
    float* orow = Out + (size_t)plane * (PLANE * PLANE)
                      + (size_t)(rowb + hi * 8) * PLANE + colb + w16 + m;
#pragma unroll
    for (int v = 0; v < 8; ++v)
        orow[v * PLANE] = acc[v];
}

extern "C" void kernel_launch(void* const* d_in, const int* in_sizes, int n_in,
                              void* d_out, int out_size, void* d_ws, size_t ws_size,
                              hipStream_t stream) {
    (void)n_in; (void)out_size; (void)d_ws; (void)ws_size;
    const float* X  = (const float*)d_in[0];
    const float* Wt = (const float*)d_in[1];
    float* Out      = (float*)d_out;

    const int planes = in_sizes[0] / (PLANE * PLANE);  // B*C = 1024
    const int blocks = planes * 32;                    // 16 trows x 2 halves
    conv3x3_wmma_f32<<<blocks, 256, 0, stream>>>(X, Wt, Out);
}